// NTXentLoss_16587163697661
// MI455X (gfx1250) — compile-verified
//
#include <hip/hip_runtime.h>
#include <math.h>

// NT-Xent loss, fused. reps is 8 MB (L2-resident); GEMM is 34.4 GFLOP fp32.
// Compute-bound -> V_WMMA_F32_16X16X4_F32 (fp32 matrix pipe, matches fp32
// reference precision). Logits bounded in [-2,2] -> plain sum-of-exp is safe.

typedef __attribute__((ext_vector_type(2))) float v2f;
typedef __attribute__((ext_vector_type(8))) float v8f;

#define B_ROWS 4096
#define D_DIM  256
#define TWOB   8192
#define N_TILES 512
#define N_SPLIT 8
#define TILES_PER_SPLIT (N_TILES / N_SPLIT)  // 64

// ---------------- row normalize: one wave (32 lanes) per row ----------------
__global__ void ntx_normalize(const float* __restrict__ z1,
                              const float* __restrict__ z2,
                              float* __restrict__ reps) {
    int wave = threadIdx.x >> 5, lane = threadIdx.x & 31;
    int row = blockIdx.x * 8 + wave;                       // grid = 1024 -> 8192 rows exact
    const float* src = (row < B_ROWS) ? (z1 + (size_t)row * D_DIM)
                                      : (z2 + (size_t)(row - B_ROWS) * D_DIM);
    float v[8];
    float ss = 0.f;
#pragma unroll
    for (int j = 0; j < 8; ++j) { v[j] = src[lane + 32 * j]; ss += v[j] * v[j]; }
#pragma unroll
    for (int m = 1; m < 32; m <<= 1) ss += __shfl_xor(ss, m, 32);
    float inv = 1.0f / fmaxf(sqrtf(ss), 1e-12f);
    float* dst = reps + (size_t)row * D_DIM;
#pragma unroll
    for (int j = 0; j < 8; ++j) dst[lane + 32 * j] = v[j] * inv;
}

// ---------------- positive logits: pos[i] = pos[i+B] = 2*dot -----------------
__global__ void ntx_pos(const float* __restrict__ reps, float* __restrict__ pos) {
    int wave = threadIdx.x >> 5, lane = threadIdx.x & 31;
    int i = blockIdx.x * 8 + wave;                         // grid = 512 -> 4096 exact
    const float* a = reps + (size_t)i * D_DIM;
    const float* b = reps + (size_t)(i + B_ROWS) * D_DIM;
    float s = 0.f;
#pragma unroll
    for (int j = 0; j < 8; ++j) s += a[lane + 32 * j] * b[lane + 32 * j];
#pragma unroll
    for (int m = 1; m < 32; m <<= 1) s += __shfl_xor(s, m, 32);
    if (lane == 0) { pos[i] = 2.0f * s; pos[i + B_ROWS] = 2.0f * s; }
}

// ------------- fused sim-tile GEMM (f32 WMMA) + exp-row-sum ------------------
// One wave owns one 16-row M-tile and 64 N-tiles. A panel lives in 128 VGPRs.
// 4 independent accumulator chains over K feed the XDL pipe; D layouts match,
// so chains sum element-wise. Per-(split,row) partial sums -> deterministic.
__global__ void __launch_bounds__(128, 1)
ntx_simexp(const float* __restrict__ reps, const float* __restrict__ pos,
           float* __restrict__ partial) {
    int wave = threadIdx.x >> 5, lane = threadIdx.x & 31;
    int g = blockIdx.x * 4 + wave;                         // grid = 1024 -> 4096 waves
    int m_tile = g >> 3, n_split = g & 7;
    int m0 = m_tile * 16;
    int half = lane >> 4, nl = lane & 15;

    // A 16x4 f32 layout: lane holds row m0+nl; VGPR0/1 = K {0,1} (half 0) / {2,3} (half 1)
    const float* Abase = reps + (size_t)(m0 + nl) * D_DIM + 2 * half;
    v2f aT[64];
#pragma unroll
    for (int kk = 0; kk < 64; ++kk) aT[kk] = *(const v2f*)(Abase + 4 * kk);

    float psum[8];
#pragma unroll
    for (int v = 0; v < 8; ++v) psum[v] = 0.f;

    for (int t = 0; t < TILES_PER_SPLIT; ++t) {
        int n0 = (n_split * TILES_PER_SPLIT + t) * 16;
        const float* Bbase = reps + (size_t)(n0 + nl) * D_DIM + 2 * half;
        v8f acc0 = {}, acc1 = {}, acc2 = {}, acc3 = {};
#pragma unroll
        for (int kk = 0; kk < 64; kk += 4) {
            v2f b0 = *(const v2f*)(Bbase + 4 * kk);
            v2f b1 = *(const v2f*)(Bbase + 4 * kk + 4);
            v2f b2 = *(const v2f*)(Bbase + 4 * kk + 8);
            v2f b3 = *(const v2f*)(Bbase + 4 * kk + 12);
            acc0 = __builtin_amdgcn_wmma_f32_16x16x4_f32(false, aT[kk + 0], false, b0,
                                                         (short)0, acc0, false, false);
            acc1 = __builtin_amdgcn_wmma_f32_16x16x4_f32(false, aT[kk + 1], false, b1,
                                                         (short)0, acc1, false, false);
            acc2 = __builtin_amdgcn_wmma_f32_16x16x4_f32(false, aT[kk + 2], false, b2,
                                                         (short)0, acc2, false, false);
            acc3 = __builtin_amdgcn_wmma_f32_16x16x4_f32(false, aT[kk + 3], false, b3,
                                                         (short)0, acc3, false, false);
        }
        v8f d = acc0 + acc1 + acc2 + acc3;
        bool diagTile = (n0 == m0);
#pragma unroll
        for (int v = 0; v < 8; ++v) {
            // D layout: VGPR v, lanes 0-15 -> M=v, N=lane; lanes 16-31 -> M=v+8, N=lane-16
            float val = d[v] * 2.0f;                        // /TEMPERATURE
            if (diagTile && (v + 8 * half) == nl) val = pos[n0 + nl];
            psum[v] += expf(val);
        }
    }
#pragma unroll
    for (int v = 0; v < 8; ++v) {
        float s = psum[v];
        s += __shfl_xor(s, 1, 32);
        s += __shfl_xor(s, 2, 32);
        s += __shfl_xor(s, 4, 32);
        s += __shfl_xor(s, 8, 32);
        if (nl == 0)  // lanes 0 & 16 write rows m0+v and m0+v+8 of this split slice
            partial[(size_t)n_split * TWOB + m0 + v + 8 * half] = s;
    }
}

// ---------------- finalize: mean(log(S_i) - pos_i) ---------------------------
__global__ void ntx_finalize(const float* __restrict__ partial,
                             const float* __restrict__ pos,
                             float* __restrict__ out) {
    __shared__ float sdata[256];
    float acc = 0.f;
    for (int i = threadIdx.x; i < TWOB; i += 256) {
        float s = 0.f;
#pragma unroll
        for (int p = 0; p < N_SPLIT; ++p) s += partial[(size_t)p * TWOB + i];
        acc += logf(s) - pos[i];
    }
    sdata[threadIdx.x] = acc;
    __syncthreads();
    for (int o = 128; o > 0; o >>= 1) {
        if ((int)threadIdx.x < o) sdata[threadIdx.x] += sdata[threadIdx.x + o];
        __syncthreads();
    }
    if (threadIdx.x == 0) out[0] = sdata[0] * (1.0f / (float)TWOB);
}

extern "C" void kernel_launch(void* const* d_in, const int* in_sizes, int n_in,
                              void* d_out, int out_size, void* d_ws, size_t ws_size,
                              hipStream_t stream) {
    (void)in_sizes; (void)n_in; (void)out_size; (void)ws_size;
    const float* z1 = (const float*)d_in[0];
    const float* z2 = (const float*)d_in[1];
    float* out = (float*)d_out;

    float* reps    = (float*)d_ws;                  // 8192*256 f32 = 8 MB
    float* pos     = reps + (size_t)TWOB * D_DIM;   // 8192 f32
    float* partial = pos + TWOB;                    // 8*8192 f32 (per-split row sums)

    ntx_normalize<<<1024, 256, 0, stream>>>(z1, z2, reps);
    ntx_pos      <<<512, 256, 0, stream>>>(reps, pos);
    ntx_simexp   <<<1024, 128, 0, stream>>>(reps, pos, partial);
    ntx_finalize <<<1, 256, 0, stream>>>(partial, pos, out);
}